// SMPL_33423435497735
// MI455X (gfx1250) — compile-verified
//
#include <hip/hip_runtime.h>
#include <stdint.h>

// SMPL forward kinematics for MI455X (gfx1250, wave32).
// B = 131072 bodies, 24 joints. ~77 MB total traffic -> ~3.3us HBM floor at
// 23.3 TB/s; ~0.4 GFLOP fp32 -> comparable VALU time. WMMA does not apply
// (4x4 chain composes, 8-deep serial dependency, not 16x16xK shaped); the
// CDNA5 feature that pays here is the async global<->LDS data mover, used
// with GVS addressing + IOFFSET (which advances BOTH the LDS and global
// address, ISA §15.18) so each 18-instruction copy reuses one LDS VGPR and
// one 32-bit offset VGPR with only the immediate varying.

#define NJ  24
#define BPB 128                 // bodies per block == threads per block (4 waves)

// LDS layout in floats
#define POSE_F      0           // BPB*69 = 8832  (linear image of body_pose chunk)
#define ORI_F       8832        // BPB*3  = 384   (linear image of global_orient chunk)
#define TRS_F       9216        // BPB*3  = 384   (linear image of transl chunk)
#define OUT_F       9600        // BPB*OUT_STRIDE (bank-padded output staging)
#define OUT_STRIDE  76          // 72 used + 4 pad: 76%64=12 -> 2-way conflicts; 304B stride is 16B aligned
#define SMEM_F      (OUT_F + BPB * OUT_STRIDE)   // 19328 floats = 77312 bytes -> 4 blocks/WGP

constexpr int kPar[NJ] = {-1, 0, 0, 0, 1, 2, 3, 4, 5, 6, 7, 8, 9, 9, 9, 12, 13, 14, 16, 17, 18, 19, 20, 21};

// ---- CDNA5 async global<->LDS, GVS mode (SGPR64 base + VGPR32 offset + imm) ----
template <int OFF>
__device__ __forceinline__ void async_g2l_b128(uint32_t lds_byte_off, uint32_t voff, uint64_t sbase) {
  asm volatile("global_load_async_to_lds_b128 %0, %1, %2 offset:%3"
               :: "v"(lds_byte_off), "v"(voff), "s"(sbase), "i"(OFF)
               : "memory");
}
template <int OFF>
__device__ __forceinline__ void async_l2g_b128(uint32_t voff, uint32_t lds_byte_off, uint64_t sbase) {
  asm volatile("global_store_async_from_lds_b128 %0, %1, %2 offset:%3"
               :: "v"(voff), "v"(lds_byte_off), "s"(sbase), "i"(OFF)
               : "memory");
}
__device__ __forceinline__ void s_wait_async0() {
  asm volatile("s_wait_asynccnt 0" ::: "memory");
}

__global__ __launch_bounds__(BPB) void smpl_fk_kernel(
    const float* __restrict__ body_pose,     // (B, 69)
    const float* __restrict__ global_orient, // (B, 3)
    const float* __restrict__ transl,        // (B, 3)
    const float* __restrict__ rest_joints,   // (24, 3) uniform -> scalar cache
    float* __restrict__ out)                 // (B, 24, 3)
{
  __shared__ __align__(16) float smem[SMEM_F];
  const int t = threadIdx.x;
  const size_t blk = blockIdx.x;
  const uint32_t smem_base = (uint32_t)(uintptr_t)(&smem[0]);  // low 32 bits of flat shared addr == LDS byte offset
  const uint32_t voff16 = (uint32_t)t * 16u;

  // ---------- stage inputs: linear async copies, one VGPR pair reused ----------
  {
    const uint64_t pbase = (uint64_t)(uintptr_t)((const char*)body_pose + blk * (size_t)(BPB * 69 * 4));
    const uint32_t lds_pose = smem_base + (uint32_t)(POSE_F * 4) + voff16;
#define POSE_LD(K) async_g2l_b128<(K) * 2048>(lds_pose, voff16, pbase)
    POSE_LD(0);  POSE_LD(1);  POSE_LD(2);  POSE_LD(3);  POSE_LD(4);  POSE_LD(5);
    POSE_LD(6);  POSE_LD(7);  POSE_LD(8);  POSE_LD(9);  POSE_LD(10); POSE_LD(11);
    POSE_LD(12); POSE_LD(13); POSE_LD(14); POSE_LD(15); POSE_LD(16);
    if (t < 32) POSE_LD(17);             // 2208 float4s total = 17*128 + 32
#undef POSE_LD
    if (t < (BPB * 3) / 4) {             // 96 float4s each
      const uint64_t obase = (uint64_t)(uintptr_t)((const char*)global_orient + blk * (size_t)(BPB * 3 * 4));
      async_g2l_b128<0>(smem_base + (uint32_t)(ORI_F * 4) + voff16, voff16, obase);
      const uint64_t tbase = (uint64_t)(uintptr_t)((const char*)transl + blk * (size_t)(BPB * 3 * 4));
      async_g2l_b128<0>(smem_base + (uint32_t)(TRS_F * 4) + voff16, voff16, tbase);
    }
  }
  s_wait_async0();     // this wave's async loads have landed in LDS
  __syncthreads();     // all waves' slices visible

  // ---------- per-body FK (one thread = one body) ----------
  const float* myPose = &smem[POSE_F + t * 69];   // stride 69 (==5 mod 64) -> conflict-free
  const float* myOri  = &smem[ORI_F + t * 3];
  float* myOut = &smem[OUT_F + t * OUT_STRIDE];

  float AR[NJ][9];   // accumulated world rotations (unrolled -> registers, liveness-pruned)
  float AT[NJ][3];   // accumulated world translations (transl folded in at the root)

#pragma unroll
  for (int i = 0; i < NJ; ++i) {
    float ax, ay, az;
    if (i == 0) { ax = myOri[0]; ay = myOri[1]; az = myOri[2]; }
    else {
      ax = myPose[(i - 1) * 3 + 0];
      ay = myPose[(i - 1) * 3 + 1];
      az = myPose[(i - 1) * 3 + 2];
    }

    // Rodrigues, matching reference: angle = |aa + 1e-8|, axis = aa / angle
    float ex = ax + 1e-8f, ey = ay + 1e-8f, ez = az + 1e-8f;
    float d2  = ex * ex + ey * ey + ez * ez;
    float inv = __builtin_amdgcn_rsqf(d2);        // 1/angle
    float ang = d2 * inv;                         // angle (one trans op serves both)
    float rx = ax * inv, ry = ay * inv, rz = az * inv;
    float s, c;
    __sincosf(ang, &s, &c);
    float C = 1.0f - c;
    // shared products: each R entry is one FMA off these
    float Crx = C * rx, Cry = C * ry, Crz = C * rz;
    float srx = s * rx, sry = s * ry, srz = s * rz;
    // diag: 1 + C*(ri^2 - n2) == 1 - C*(rj^2 + rk^2)  (n2 = |axis|^2, K@K form)
    float R0 = 1.0f - (Cry * ry + Crz * rz);
    float R4 = 1.0f - (Crx * rx + Crz * rz);
    float R8 = 1.0f - (Crx * rx + Cry * ry);
    float R1 = Crx * ry - srz;
    float R2 = Crx * rz + sry;
    float R3 = Cry * rx + srz;
    float R5 = Cry * rz - srx;
    float R6 = Crz * rx - sry;
    float R7 = Crz * ry + srx;

    const int p = kPar[i];
    float jx, jy, jz;
    if (i == 0) {
      // fold transl into the root translation: it propagates to every joint
      jx = rest_joints[0] + smem[TRS_F + t * 3 + 0];
      jy = rest_joints[1] + smem[TRS_F + t * 3 + 1];
      jz = rest_joints[2] + smem[TRS_F + t * 3 + 2];
    } else {
      jx = rest_joints[i * 3 + 0] - rest_joints[p * 3 + 0];
      jy = rest_joints[i * 3 + 1] - rest_joints[p * 3 + 1];
      jz = rest_joints[i * 3 + 2] - rest_joints[p * 3 + 2];
    }

    if (i == 0) {
      AR[0][0] = R0; AR[0][1] = R1; AR[0][2] = R2;
      AR[0][3] = R3; AR[0][4] = R4; AR[0][5] = R5;
      AR[0][6] = R6; AR[0][7] = R7; AR[0][8] = R8;
      AT[0][0] = jx; AT[0][1] = jy; AT[0][2] = jz;
    } else {
      AR[i][0] = AR[p][0] * R0 + AR[p][1] * R3 + AR[p][2] * R6;
      AR[i][1] = AR[p][0] * R1 + AR[p][1] * R4 + AR[p][2] * R7;
      AR[i][2] = AR[p][0] * R2 + AR[p][1] * R5 + AR[p][2] * R8;
      AR[i][3] = AR[p][3] * R0 + AR[p][4] * R3 + AR[p][5] * R6;
      AR[i][4] = AR[p][3] * R1 + AR[p][4] * R4 + AR[p][5] * R7;
      AR[i][5] = AR[p][3] * R2 + AR[p][4] * R5 + AR[p][5] * R8;
      AR[i][6] = AR[p][6] * R0 + AR[p][7] * R3 + AR[p][8] * R6;
      AR[i][7] = AR[p][6] * R1 + AR[p][7] * R4 + AR[p][8] * R7;
      AR[i][8] = AR[p][6] * R2 + AR[p][7] * R5 + AR[p][8] * R8;
      AT[i][0] = AR[p][0] * jx + AR[p][1] * jy + AR[p][2] * jz + AT[p][0];
      AT[i][1] = AR[p][3] * jx + AR[p][4] * jy + AR[p][5] * jz + AT[p][1];
      AT[i][2] = AR[p][6] * jx + AR[p][7] * jy + AR[p][8] * jz + AT[p][2];
    }

    // AT already includes transl (seeded at the root) -> direct store
    myOut[i * 3 + 0] = AT[i][0];
    myOut[i * 3 + 1] = AT[i][1];
    myOut[i * 3 + 2] = AT[i][2];
  }

  __syncthreads();   // out staging fully written (barrier waits dscnt)

  // ---------- async store: per-thread-contiguous, padded LDS -> dense global ----------
  // Thread t stores body t's 18 float4s: global advances 16B/iter (voff=t*288),
  // padded LDS also advances 16B/iter (vsrc = OUT + t*304). L2 (192 MB) merges
  // the 16B sectors into full lines before writeback.
  {
    const uint64_t sbase_out = (uint64_t)(uintptr_t)(out + blk * (size_t)(BPB * 72));
    const uint32_t voff_out  = (uint32_t)t * 288u;
    const uint32_t lds_out   = smem_base + (uint32_t)(OUT_F * 4) + (uint32_t)t * (OUT_STRIDE * 4);
#define OUT_ST(K) async_l2g_b128<(K) * 16>(voff_out, lds_out, sbase_out)
    OUT_ST(0);  OUT_ST(1);  OUT_ST(2);  OUT_ST(3);  OUT_ST(4);  OUT_ST(5);
    OUT_ST(6);  OUT_ST(7);  OUT_ST(8);  OUT_ST(9);  OUT_ST(10); OUT_ST(11);
    OUT_ST(12); OUT_ST(13); OUT_ST(14); OUT_ST(15); OUT_ST(16); OUT_ST(17);
#undef OUT_ST
  }
  s_wait_async0();   // belt-and-braces; S_ENDPGM also implies wait-idle
}

extern "C" void kernel_launch(void* const* d_in, const int* in_sizes, int n_in,
                              void* d_out, int out_size, void* d_ws, size_t ws_size,
                              hipStream_t stream) {
  (void)n_in; (void)out_size; (void)d_ws; (void)ws_size;
  const float* body_pose     = (const float*)d_in[0];
  const float* global_orient = (const float*)d_in[1];
  const float* transl        = (const float*)d_in[2];
  const float* rest_joints   = (const float*)d_in[3];
  float* out = (float*)d_out;

  const int B = in_sizes[0] / 69;       // 131072
  const int blocks = B / BPB;           // 1024 (B is a multiple of 128)
  smpl_fk_kernel<<<blocks, BPB, 0, stream>>>(body_pose, global_orient, transl,
                                             rest_joints, out);
}